// InteractionGNNAggregator_79173427134506
// MI455X (gfx1250) — compile-verified
//
#include <hip/hip_runtime.h>
#include <hip/hip_bf16.h>
#include <math.h>

// ---------------------------------------------------------------------------
// InteractionGNNAggregator for MI455X (gfx1250, wave32, WMMA).
// Heavy GEMMs use V_WMMA_F32_16X16X4_F32 (fp32, matches reference precision).
// ---------------------------------------------------------------------------

typedef __attribute__((ext_vector_type(2))) float v2f;
typedef __attribute__((ext_vector_type(8))) float v8f;

#define B_   128
#define L_   50
#define N_   12
#define N2_  12
#define S_   20
#define M_   12
#define D_   100
#define KT   25      // K-steps of 4 for D=100
#define NT   7       // N tiles of 16 covering 112 (>=100, zero padded)
#define NPAD 112

// ===========================================================================
// Fused score kernel, K = 100:
//   x[row]   = A[row / a_div] (elementwise *) Bm[row]         (D floats)
//   y        = leaky_relu(x @ W, 0.2)                         (D floats)
//   score    = y . v
// 4 waves/block, 16 rows per wave, 64 rows per block.
// W staged transposed in LDS: Wl[n*D_ + k] (columns 100..111 zeroed).
// ===========================================================================
__global__ __launch_bounds__(128)
void score_d100_kernel(const float* __restrict__ A,   // [RA, D]
                       const float* __restrict__ Bm,  // [R, D]
                       const float* __restrict__ W,   // [D, D] row-major (K x N)
                       const float* __restrict__ v,   // [D]
                       float* __restrict__ scores,    // [R]
                       int R, int a_div)
{
    __shared__ float Wl[NPAD * D_];   // transposed, column-major-by-n
    __shared__ float vl[NPAD];

    const int tid = threadIdx.x;
    for (int idx = tid; idx < NPAD * D_; idx += 128) {
        int n = idx / D_, k = idx - n * D_;
        Wl[idx] = (n < D_) ? W[k * D_ + n] : 0.0f;
    }
    for (int idx = tid; idx < NPAD; idx += 128)
        vl[idx] = (idx < D_) ? v[idx] : 0.0f;
    __syncthreads();

    const int wave = tid >> 5;
    const int lane = tid & 31;
    const int li   = lane & 15;
    const int hi   = lane >> 4;                // which K-pair (0 -> k,k+1 ; 1 -> k+2,k+3)
    const int tile = blockIdx.x * 64 + wave * 16;
    if (tile >= R) return;                     // grids are exact; wave-uniform anyway

    const int row  = tile + li;
    const int arow = row / a_div;

    const v2f* brow = (const v2f*)(Bm + (size_t)row  * D_);
    const v2f* arw  = (const v2f*)(A  + (size_t)arow * D_);

    // A fragments: lane (li,hi) holds x[row][4*kk + 2*hi + {0,1}]
    v2f af[KT];
#pragma unroll
    for (int kk = 0; kk < KT; ++kk)
        af[kk] = arw[kk * 2 + hi] * brow[kk * 2 + hi];

    float part[8];
#pragma unroll
    for (int r = 0; r < 8; ++r) part[r] = 0.0f;

    for (int nt = 0; nt < NT; ++nt) {
        v8f acc = {0.f, 0.f, 0.f, 0.f, 0.f, 0.f, 0.f, 0.f};
        const v2f* wnv = (const v2f*)&Wl[(nt * 16 + li) * D_];  // column nt*16+li
#pragma unroll
        for (int kk = 0; kk < KT; ++kk) {
            acc = __builtin_amdgcn_wmma_f32_16x16x4_f32(
                false, af[kk], false, wnv[kk * 2 + hi], (short)0, acc, false, false);
        }
        const float w2n = vl[nt * 16 + li];     // column weight (fixed per lane)
#pragma unroll
        for (int r = 0; r < 8; ++r) {
            float y = acc[r];
            y = (y > 0.f) ? y : 0.2f * y;       // leaky_relu(0.2)
            part[r] += y * w2n;
        }
    }

    // Reduce over the 16 lanes of each half (columns); rows m = r + 8*hi.
#pragma unroll
    for (int m = 1; m < 16; m <<= 1) {
#pragma unroll
        for (int r = 0; r < 8; ++r)
            part[r] += __shfl_xor(part[r], m, 32);
    }
    if (li < 8) {
        float val = part[0];
#pragma unroll
        for (int r = 1; r < 8; ++r)
            if (li == r) val = part[r];
        scores[tile + hi * 8 + li] = val;
    }
}

// ===========================================================================
// Fused target score kernel, K = 200 (two 100-K phases):
//   x = [ T[b,s] * TN[b,s,m] , SL[b,(l=0)] + TN[b,s,m] ]   (2D floats)
//   score = tanh(x @ W3) . v4
// ===========================================================================
__global__ __launch_bounds__(128)
void tar_score_kernel(const float* __restrict__ T,    // [B,S,D]
                      const float* __restrict__ TN,   // [B,S,M,D]
                      const float* __restrict__ SL,   // [B,L,D]; row l=0 used
                      const float* __restrict__ W3,   // [2D, D]
                      const float* __restrict__ v4,   // [D]
                      float* __restrict__ scores)     // [B*S*M]
{
    __shared__ float Wl[NPAD * D_];
    __shared__ float vl[NPAD];

    const int tid = threadIdx.x;
    for (int idx = tid; idx < NPAD; idx += 128)
        vl[idx] = (idx < D_) ? v4[idx] : 0.0f;

    const int wave = tid >> 5;
    const int lane = tid & 31;
    const int li   = lane & 15;
    const int hi   = lane >> 4;
    const int row  = blockIdx.x * 64 + wave * 16 + li;   // (b*S+s)*M + m
    const int bs   = row / M_;
    const int b    = bs / S_;

    const v2f* tn = (const v2f*)(TN + (size_t)row * D_);
    const v2f* tt = (const v2f*)(T  + (size_t)bs  * D_);
    const v2f* sl = (const v2f*)(SL + (size_t)b * (L_ * D_));  // l = 0 row

    v8f acc[NT];
#pragma unroll
    for (int nt = 0; nt < NT; ++nt)
        acc[nt] = (v8f){0.f, 0.f, 0.f, 0.f, 0.f, 0.f, 0.f, 0.f};

    for (int phase = 0; phase < 2; ++phase) {
        __syncthreads();   // previous Wl consumers done
        for (int idx = tid; idx < NPAD * D_; idx += 128) {
            int n = idx / D_, k = idx - n * D_;
            Wl[idx] = (n < D_) ? W3[(phase * D_ + k) * D_ + n] : 0.0f;
        }
        __syncthreads();

        v2f af[KT];
#pragma unroll
        for (int kk = 0; kk < KT; ++kk) {
            v2f b2 = tn[kk * 2 + hi];
            af[kk] = (phase == 0) ? (tt[kk * 2 + hi] * b2)
                                  : (sl[kk * 2 + hi] + b2);
        }
        for (int nt = 0; nt < NT; ++nt) {
            const v2f* wnv = (const v2f*)&Wl[(nt * 16 + li) * D_];
#pragma unroll
            for (int kk = 0; kk < KT; ++kk) {
                acc[nt] = __builtin_amdgcn_wmma_f32_16x16x4_f32(
                    false, af[kk], false, wnv[kk * 2 + hi], (short)0, acc[nt], false, false);
            }
        }
    }

    float part[8];
#pragma unroll
    for (int r = 0; r < 8; ++r) part[r] = 0.0f;
#pragma unroll
    for (int nt = 0; nt < NT; ++nt) {
        const float w4n = vl[nt * 16 + li];
#pragma unroll
        for (int r = 0; r < 8; ++r)
            part[r] += tanhf(acc[nt][r]) * w4n;
    }
#pragma unroll
    for (int m = 1; m < 16; m <<= 1) {
#pragma unroll
        for (int r = 0; r < 8; ++r)
            part[r] += __shfl_xor(part[r], m, 32);
    }
    if (li < 8) {
        float val = part[0];
#pragma unroll
        for (int r = 1; r < 8; ++r)
            if (li == r) val = part[r];
        scores[blockIdx.x * 64 + wave * 16 + hi * 8 + li] = val;
    }
}

// ===========================================================================
// Softmax over nn neighbors + weighted sum (+ optional add/scale + l2norm).
// One block of 128 threads per group g; thread d < 100 owns feature d.
// ===========================================================================
__global__ __launch_bounds__(128)
void agg_kernel(const float* __restrict__ sc,   // [G, nn]
                const float* __restrict__ NB,   // [G*nn, D]
                const float* __restrict__ addv, // [G, D] or nullptr
                float* __restrict__ out,        // [G, D]
                int nn, float scale, int do_norm)
{
    __shared__ float att[16];
    __shared__ float red[128];
    const int g = blockIdx.x;
    const int tid = threadIdx.x;

    if (tid == 0) {
        float mx = -1e30f;
        for (int n = 0; n < nn; ++n) mx = fmaxf(mx, sc[(size_t)g * nn + n]);
        float s = 0.f;
        for (int n = 0; n < nn; ++n) {
            float e = __expf(sc[(size_t)g * nn + n] - mx);
            att[n] = e; s += e;
        }
        float inv = 1.f / s;
        for (int n = 0; n < nn; ++n) att[n] *= inv;
    }
    __syncthreads();

    float tot = 0.f;
    if (tid < D_) {
        const float* nb = NB + (size_t)g * nn * D_ + tid;
        for (int n = 0; n < nn; ++n)
            tot += att[n] * nb[(size_t)n * D_];
        if (addv) tot = (tot + addv[(size_t)g * D_ + tid]) * scale;
    }

    if (do_norm) {
        red[tid] = (tid < D_) ? tot * tot : 0.f;
        __syncthreads();
        for (int s = 64; s > 0; s >>= 1) {
            if (tid < s) red[tid] += red[tid + s];
            __syncthreads();
        }
        float nrm = fmaxf(sqrtf(red[0]), 1e-12f);
        if (tid < D_) out[(size_t)g * D_ + tid] = tot / nrm;
    } else {
        if (tid < D_) out[(size_t)g * D_ + tid] = tot;
    }
}

// ===========================================================================
// Output assembly (float4-vectorized; D=100 -> 25 float4 per row).
// sess stack: [3,B,S,L,D], layer l broadcast over S from [B,L,D] sources.
// ===========================================================================
__global__ void sess_out_kernel(const float4* __restrict__ s0,
                                const float4* __restrict__ s1,
                                const float4* __restrict__ s2,
                                float4* __restrict__ out)
{
    int idx = blockIdx.x * blockDim.x + threadIdx.x;   // < 3*B*S*L*25
    int d4 = idx % 25;  int q = idx / 25;
    int l  = q % L_;    q /= L_;
    /* s */             q /= S_;                        // s unused (broadcast)
    int b  = q % B_;    int layer = q / B_;
    const float4* src = (layer == 0) ? s0 : ((layer == 1) ? s1 : s2);
    out[idx] = src[(b * L_ + l) * 25 + d4];
}

// tar stack: 10 slots [t0,t1,t0,t1,t2] x2, each [B,S,D].
__global__ void tar_out_kernel(const float4* __restrict__ t0,
                               const float4* __restrict__ t1,
                               const float4* __restrict__ t2,
                               float4* __restrict__ out)
{
    int idx = blockIdx.x * blockDim.x + threadIdx.x;   // < 10*B*S*25
    const int per = B_ * S_ * 25;
    int slot = idx / per;
    int inner = idx - slot * per;
    int j = slot % 5;                                  // pattern repeats every 5
    const float4* src = (j == 4) ? t2 : ((j & 1) ? t1 : t0);
    out[idx] = src[inner];
}

// ===========================================================================
// Host-side orchestration
// ===========================================================================
extern "C" void kernel_launch(void* const* d_in, const int* in_sizes, int n_in,
                              void* d_out, int out_size, void* d_ws, size_t ws_size,
                              hipStream_t stream) {
    const float* sess_self  = (const float*)d_in[0];   // [B,L,D]
    const float* sess_nb    = (const float*)d_in[1];   // [B,L,N,D]
    const float* sess_nb2   = (const float*)d_in[2];   // [B,L,N,N2,D]
    const float* tar_self   = (const float*)d_in[3];   // [B,S,D]
    const float* tar_nb     = (const float*)d_in[4];   // [B,S,M,D]
    const float* w1         = (const float*)d_in[5];   // [2,D,D]
    const float* w2         = (const float*)d_in[6];   // [2,D,1]
    const float* w3         = (const float*)d_in[7];   // [2,2D,D]
    const float* w4         = (const float*)d_in[8];   // [2,D,1]
    // d_in[9] = session_len (unused by the reference forward)

    // Workspace layout (floats); all offsets multiples of 4 -> float4 safe.
    float* wsf  = (float*)d_ws;
    float* emb1 = wsf;                                  // B*L*D   = 640000
    float* emb2 = emb1 + B_ * L_ * D_;                  // 640000
    float* snu  = emb2 + B_ * L_ * D_;                  // B*L*N*D = 7680000
    float* t1   = snu  + (size_t)B_ * L_ * N_ * D_;     // B*S*D   = 256000
    float* t2   = t1   + B_ * S_ * D_;                  // 256000
    float* scS  = t2   + B_ * S_ * D_;                  // B*L*N   = 76800
    float* sc2  = scS  + B_ * L_ * N_;                  // B*L*N*N2= 921600
    float* scT  = sc2  + (size_t)B_ * L_ * N_ * N2_;    // B*S*M   = 30720

    const int RS = B_ * L_ * N_;           // 76800  (mult of 64)
    const int R2 = B_ * L_ * N_ * N2_;     // 921600 (mult of 64)
    const int RT = B_ * S_ * M_;           // 30720  (mult of 64)

    // ---- layer 0: session attention (mean of broadcast == sess_self) ----
    score_d100_kernel<<<RS / 64, 128, 0, stream>>>(
        sess_self, sess_nb, w1, w2, scS, RS, N_);
    agg_kernel<<<B_ * L_, 128, 0, stream>>>(
        scS, sess_nb, nullptr, emb1, N_, 1.0f, 1);

    // ---- 2-hop aggregation (layer 0 weights, original sess_neighbor) ----
    score_d100_kernel<<<R2 / 64, 128, 0, stream>>>(
        sess_nb, sess_nb2, w1, w2, sc2, R2, N2_);
    agg_kernel<<<B_ * L_ * N_, 128, 0, stream>>>(
        sc2, sess_nb2, nullptr, snu, N2_, 1.0f, 0);

    // ---- layer 0: target attention (sess_last = emb1[:, l=0, :]) ----
    tar_score_kernel<<<RT / 64, 128, 0, stream>>>(
        tar_self, tar_nb, emb1, w3, w4, scT);
    agg_kernel<<<B_ * S_, 128, 0, stream>>>(
        scT, tar_nb, tar_self, t1, M_, 0.5f, 1);

    // ---- layer 1: session attention (mean == emb1, neighbors = snu) ----
    score_d100_kernel<<<RS / 64, 128, 0, stream>>>(
        emb1, snu, w1 + D_ * D_, w2 + D_, scS, RS, N_);
    agg_kernel<<<B_ * L_, 128, 0, stream>>>(
        scS, snu, nullptr, emb2, N_, 1.0f, 1);

    // ---- layer 1: target attention (t = t1, sess_last = sess_self[:,0,:]) ----
    tar_score_kernel<<<RT / 64, 128, 0, stream>>>(
        t1, tar_nb, sess_self, w3 + 2 * D_ * D_, w4 + D_, scT);
    agg_kernel<<<B_ * S_, 128, 0, stream>>>(
        scT, tar_nb, tar_self, t2, M_, 0.5f, 1);

    // ---- assemble outputs ----
    float* out_sess = (float*)d_out;                          // [3,B,S,L,D]
    float* out_tar  = out_sess + (size_t)3 * B_ * S_ * L_ * D_;  // [10,B,S,D]

    const int sessV4 = 3 * B_ * S_ * L_ * 25;   // 9,600,000 (mult of 256)
    sess_out_kernel<<<sessV4 / 256, 256, 0, stream>>>(
        (const float4*)sess_self, (const float4*)emb1, (const float4*)emb2,
        (float4*)out_sess);

    const int tarV4 = 10 * B_ * S_ * 25;        // 640,000 (mult of 256)
    tar_out_kernel<<<tarV4 / 256, 256, 0, stream>>>(
        (const float4*)tar_self, (const float4*)t1, (const float4*)t2,
        (float4*)out_tar);

    (void)in_sizes; (void)n_in; (void)out_size; (void)ws_size;
}